// Custom_CE_Loss_74792560492734
// MI455X (gfx1250) — compile-verified
//
#include <hip/hip_runtime.h>
#include <math.h>

#define T_DIM 4096
#define B_DIM 512
#define NSLOT 8          // LDS ring slots (steps of HBM latency cover)
#define PFD   16         // scalar-stream prefetch distance (steps)

// ---- CDNA5 async-to-LDS primitives (inline asm; CDNA5 ISA) ----
// LDS byte address for the async op: for a generic pointer to __shared__ data,
// the flat address maps to LDS via addr[31:0] (ISA 10.2 aperture rule), so a
// 32-bit truncation of the generic pointer is the LDS address VGPR value.
__device__ __forceinline__ void async_b128(const float* g, float* l) {
#if defined(__AMDGCN__)
  unsigned lds_addr = (unsigned)(unsigned long long)l;
  asm volatile("global_load_async_to_lds_b128 %0, %1, off"
               :
               : "v"(lds_addr), "v"(g)
               : "memory");
#else
  ((float4*)l)[0] = ((const float4*)g)[0];   // host parse only; never executed
#endif
}

#if defined(__AMDGCN__)
#define WAIT_ASYNC_STEADY() asm volatile("s_wait_asynccnt 0x15" ::: "memory") // (NSLOT-1)*3 = 21
#define WAIT_ASYNC_ALL()    asm volatile("s_wait_asynccnt 0x0"  ::: "memory")
#define WAIT_DS0()          asm volatile("s_wait_dscnt 0x0"     ::: "memory")
#else
#define WAIT_ASYNC_STEADY()
#define WAIT_ASYNC_ALL()
#define WAIT_DS0()
#endif

// Stage one lane's 12-float logits row (48B) into LDS asynchronously.
__device__ __forceinline__ void stage_logits_row(const float* __restrict__ g, float* l) {
  async_b128(g + 0, l + 0);
  async_b128(g + 4, l + 4);
  async_b128(g + 8, l + 8);
}

// ---------------- Pass 2 (placed first so the disasm snippet shows it):
// the serial simulation, one wave32 per block spread across WGPs ----------------
__global__ __launch_bounds__(32)
void sim_kernel(const float* __restrict__ labels, const float* __restrict__ logits,
                const float* __restrict__ u1p, const float* __restrict__ u2p,
                const float* __restrict__ u3p, const float* __restrict__ u4p,
                const float* __restrict__ mxp, const float* __restrict__ mnp,
                const float* __restrict__ hoverPtr, float* __restrict__ partials) {
  const int lane = threadIdx.x;
  const int gb   = blockIdx.x * 32 + lane;   // batch column, 16*32 = 512

  __shared__ __align__(16) float lg[NSLOT][32 * 12];

  const float tau  = 0.005f;
  const float mB   = 1.2f;
  const float grav = 9.81f;
  const float IRzz = 1.0e-4f;
  const float hover = hoverPtr[0];

  // live state (dead states xx/xy/xd/yd eliminated)
  float z = 0.0f, q0 = 1.0f, q1 = 0.0f, q2 = 0.0f, q3 = 0.0f;
  float zd = 0.0f, pv = 0.0f, qv = 0.0f, rv = 0.0f;
  float w1 = hover, w2 = hover, w3 = hover, w4 = hover;
  float wd1 = 0.0f, wd2 = 0.0f, wd3 = 0.0f, wd4 = 0.0f;

  float l0  = labels[gb];        // t=0 term: (0 - labels[0,b])^2
  float acc = l0 * l0;

  struct Sc { float lab, u1, u2, u3, u4, mn, mx; };
  auto loadSc = [&](int t) -> Sc {
    size_t i = (size_t)t * B_DIM + gb;
    Sc s;
    s.lab = labels[i];
    s.u1 = u1p[i]; s.u2 = u2p[i]; s.u3 = u3p[i]; s.u4 = u4p[i];
    s.mn = mnp[i]; s.mx = mxp[i];
    return s;
  };

  auto do_step = [&](int slot, const Sc& sc) {
    const float4* lp = (const float4*)&lg[slot][lane * 12];
    float4 La = lp[0], Lb = lp[1], Lc = lp[2];
    // scale(k, base) = (1 + (0.5 - L[k]) * 0.95) * base
    float dxm  = (1.0f + (0.5f - La.x) * 0.95f) * 0.16f;
    float dym  = (1.0f + (0.5f - La.y) * 0.95f) * 0.16f;
    float IBxx = (1.0f + (0.5f - La.w) * 0.95f) * 0.0123f;
    float IByy = (1.0f + (0.5f - Lb.x) * 0.95f) * 0.0123f;
    float IBzz = (1.0f + (0.5f - Lb.y) * 0.95f) * 0.0123f;
    float Cd   = (1.0f + (0.5f - Lb.z) * 0.95f) * 0.1f;
    float kTh  = (1.0f + (0.5f - Lb.w) * 0.95f) * 1.076e-5f;
    float kTo  = (1.0f + (0.5f - Lc.x) * 0.95f) * 1.632e-7f;
    float tau2 = (1.0f + (0.5f - Lc.y) * 0.95f) * 0.015f;
    float kp   = (1.0f + (0.5f - Lc.z) * 0.95f) * 1.0f;
    float damp = (1.0f + (0.5f - Lc.w) * 0.95f) * 1.0f;

    float kpot = kp / (tau2 * tau2);
    float d2t  = -2.0f * damp * tau2;
    float wdd1 = d2t * wd1 - w1 + kpot * sc.u1;
    float wdd2 = d2t * wd2 - w2 + kpot * sc.u2;
    float wdd3 = d2t * wd3 - w3 + kpot * sc.u3;
    float wdd4 = d2t * wd4 - w4 + kpot * sc.u4;

    float w1c = fminf(fmaxf(w1, sc.mn), sc.mx);
    float w2c = fminf(fmaxf(w2, sc.mn), sc.mx);
    float w3c = fminf(fmaxf(w3, sc.mn), sc.mx);
    float w4c = fminf(fmaxf(w4, sc.mn), sc.mx);

    float s1 = w1c * w1c, s2 = w2c * w2c, s3 = w3c * w3c, s4 = w4c * w4c;
    float T1 = kTh * s1, T2 = kTh * s2, T3 = kTh * s3, T4 = kTh * s4;
    float Thr = T1 + T2 + T3 + T4;

    float qsq = q0 * q0 - q1 * q1 - q2 * q2 + q3 * q3;
    float nzd = zd + tau * (-Cd * zd * fabsf(zd) + Thr * qsq - grav * mB) / mB;

    float nq0 = q0 + tau * (-0.5f * (pv * q1 + qv * q2 + rv * q3));
    float nq1 = q1 + tau * ( 0.5f * (pv * q0 - qv * q3 + rv * q2));
    float nq2 = q2 + tau * ( 0.5f * (pv * q3 + qv * q0 - rv * q1));
    float nq3 = q3 + tau * ( 0.5f * (-pv * q2 + qv * q1 + rv * q0));

    float wsum = w1c - w2c + w3c - w4c;
    float npv = pv + tau * ((IByy - IBzz) * qv * rv - IRzz * wsum * qv + (T1 - T2 - T3 + T4) * dym) / IBxx;
    float nqv = qv + tau * ((IBzz - IBxx) * pv * rv + IRzz * wsum * pv + (T1 + T2 - T3 - T4) * dxm) / IByy;
    float nrv = rv + tau * ((IBxx - IByy) * pv * qv + kTo * (-s1 + s2 - s3 + s4)) / IBzz;

    float nzraw = z + tau * nzd;
    float nw1 = w1c + tau * wd1, nw2 = w2c + tau * wd2;
    float nw3 = w3c + tau * wd3, nw4 = w4c + tau * wd4;
    float nwd1 = wd1 + tau * wdd1, nwd2 = wd2 + tau * wdd2;
    float nwd3 = wd3 + tau * wdd3, nwd4 = wd4 + tau * wdd4;

    // isfinite + max_inc gate (NaN compares false -> keep old z, matching jnp.where)
    bool ok = (fabsf(nzraw) <= 3.402823466e+38f) && (fabsf(nzraw - z) <= 400.0f);
    float nz = ok ? nzraw : z;

    float e = nz - sc.lab;
    acc = fmaf(e, e, acc);

    z = nz; q0 = nq0; q1 = nq1; q2 = nq2; q3 = nq3;
    zd = nzd; pv = npv; qv = nqv; rv = nrv;
    w1 = nw1; w2 = nw2; w3 = nw3; w4 = nw4;
    wd1 = nwd1; wd2 = nwd2; wd3 = nwd3; wd4 = nwd4;
  };

  // prologue: fill the async ring with steps t = 1..NSLOT
  for (int s = 0; s < NSLOT; ++s) {
    int tt = 1 + s;
    stage_logits_row(logits + ((size_t)tt * B_DIM + gb) * 12, &lg[s][lane * 12]);
  }

  Sc cur = loadSc(1);

  // steady state: always refills; constant-immediate asynccnt throttle
  const int tmain = T_DIM - NSLOT;   // first iteration without refill
  for (int t = 1; t < tmain; ++t) {
    int slot = (t - 1) & (NSLOT - 1);
    WAIT_ASYNC_STEADY();             // oldest slot's 3 async loads complete (in-order)
    Sc nxt = loadSc(t + 1);
    if (t + PFD < T_DIM) {
      size_t pidx = (size_t)(t + PFD) * B_DIM + gb;
      __builtin_prefetch(labels + pidx, 0, 1);
      __builtin_prefetch(u1p + pidx, 0, 1);
      __builtin_prefetch(u2p + pidx, 0, 1);
      __builtin_prefetch(u3p + pidx, 0, 1);
      __builtin_prefetch(u4p + pidx, 0, 1);
      __builtin_prefetch(mnp + pidx, 0, 1);
      __builtin_prefetch(mxp + pidx, 0, 1);
      __builtin_prefetch(logits + pidx * 12, 0, 1);
    }
    do_step(slot, cur);
    WAIT_DS0();                      // slot reads done before async rewrite of same slot
    stage_logits_row(logits + ((size_t)(t + NSLOT) * B_DIM + gb) * 12, &lg[slot][lane * 12]);
    cur = nxt;
  }
  // drain: last NSLOT steps, no refills
  for (int t = tmain; t < T_DIM; ++t) {
    int slot = (t - 1) & (NSLOT - 1);
    WAIT_ASYNC_ALL();
    Sc nxt = (t + 1 < T_DIM) ? loadSc(t + 1) : cur;
    do_step(slot, cur);
    cur = nxt;
  }

  // wave32 reduction of per-column partial sums
  for (int off = 16; off > 0; off >>= 1) acc += __shfl_down(acc, off, 32);
  if (lane == 0) partials[blockIdx.x] = acc;
}

// ---------------- Pass 1: deterministic mean of logits[:,:,7] -> hover ----------------
__global__ __launch_bounds__(256) void reduce7_kernel(const float* __restrict__ logits,
                                                      float* __restrict__ partials) {
  __shared__ float sm[256];
  const int tid = threadIdx.x;
  const size_t base = (size_t)blockIdx.x * 8192;   // 256 blocks * 8192 = T*B
  float s = 0.0f;
  for (int k = 0; k < 32; ++k) {
    size_t row = base + (size_t)tid + (size_t)k * 256;
    s += logits[row * 12 + 7];
  }
  sm[tid] = s; __syncthreads();
  for (int off = 128; off > 0; off >>= 1) {
    if (tid < off) sm[tid] += sm[tid + off];
    __syncthreads();
  }
  if (tid == 0) partials[blockIdx.x] = sm[0];
}

__global__ __launch_bounds__(256) void hover_kernel(const float* __restrict__ partials,
                                                    float* __restrict__ hoverOut) {
  __shared__ float sm[256];
  const int tid = threadIdx.x;
  sm[tid] = partials[tid]; __syncthreads();
  for (int off = 128; off > 0; off >>= 1) {
    if (tid < off) sm[tid] += sm[tid + off];
    __syncthreads();
  }
  if (tid == 0) {
    float meanL7  = sm[0] / (4096.0f * 512.0f);
    float kThMean = (1.0f + (0.5f - meanL7) * 0.95f) * 1.076e-5f;
    float v = 1.2f * 9.81f / (4.0f * kThMean + 1e-12f);
    hoverOut[0] = sqrtf(fmaxf(v, 1e-6f));
  }
}

// ---------------- Pass 3: final mean ----------------
__global__ __launch_bounds__(32) void finish_kernel(const float* __restrict__ partials,
                                                    float* __restrict__ out) {
  int lane = threadIdx.x;
  float v = (lane < 16) ? partials[lane] : 0.0f;
  for (int off = 16; off > 0; off >>= 1) v += __shfl_down(v, off, 32);
  if (lane == 0) out[0] = v / (4096.0f * 512.0f);
}

extern "C" void kernel_launch(void* const* d_in, const int* in_sizes, int n_in,
                              void* d_out, int out_size, void* d_ws, size_t ws_size,
                              hipStream_t stream) {
  (void)in_sizes; (void)n_in; (void)out_size; (void)ws_size;
  const float* labels = (const float*)d_in[0];
  const float* logits = (const float*)d_in[1];
  const float* u1     = (const float*)d_in[2];
  const float* u2     = (const float*)d_in[3];
  const float* u3     = (const float*)d_in[4];
  const float* u4     = (const float*)d_in[5];
  const float* mx     = (const float*)d_in[6];  // maxMotor precedes minMotor in input order
  const float* mn     = (const float*)d_in[7];

  float* wsf      = (float*)d_ws;
  float* partials = wsf;         // 256 floats
  float* hover    = wsf + 256;   // 1 float
  float* simp     = wsf + 257;   // 16 floats

  reduce7_kernel<<<256, 256, 0, stream>>>(logits, partials);
  hover_kernel<<<1, 256, 0, stream>>>(partials, hover);
  sim_kernel<<<16, 32, 0, stream>>>(labels, logits, u1, u2, u3, u4, mx, mn, hover, simp);
  finish_kernel<<<1, 32, 0, stream>>>(simp, (float*)d_out);
}